// SublayerConnection_79370995630690
// MI455X (gfx1250) — compile-verified
//
#include <hip/hip_runtime.h>
#include <hip/hip_bf16.h>

typedef __attribute__((ext_vector_type(2))) float v2f;
typedef __attribute__((ext_vector_type(8))) float v8f;

#define SIZE_D 1024
#define SEQ_N  4096
#define BS_N   4
#define ROWS   (SEQ_N * BS_N)   // 16384 rows of length 1024, row r = s*BS + b
#define NDROP  409              // int(4096 * 0.1)

// ---------------------------------------------------------------------------
// Kernel 1: w[r] = dot(h_row[r], e) via V_WMMA_F32_16X16X4_F32.
// One wave owns 16 consecutive rows (M dim). B columns all hold the same
// embedding chunk, so C[m, n] == row-m dot product for every n; lanes 0 and 16
// (N=0) write out M=0..7 / M=8..15 from their 8 accumulator VGPRs.
// A 16x4 f32 layout: lanes 0-15 hold {K0,K1}, lanes 16-31 hold {K2,K3};
// B (4x16) mirrors it per lane-half, so both operands are v2f loads at
// (k + 4j + 2*half).
// ---------------------------------------------------------------------------
__global__ __launch_bounds__(256) void task_attn_gemv_wmma(
    const float* __restrict__ h, const float* __restrict__ e,
    float* __restrict__ wdot) {
  const int lane = threadIdx.x & 31;
  const int wave = blockIdx.x * 8 + (threadIdx.x >> 5);
  const int r0   = wave * 16;            // 1024 waves cover 16384 rows
  const int half = lane >> 4;            // 0: K0/K1 pairs, 1: K2/K3 pairs
  const int mrow = lane & 15;

  const float* rowp = h + (size_t)(r0 + mrow) * SIZE_D + half * 2;
  const float* ep   = e + half * 2;

  v8f acc = {0.f, 0.f, 0.f, 0.f, 0.f, 0.f, 0.f, 0.f};

  for (int k = 0; k < SIZE_D; k += 16) {
    __builtin_prefetch(rowp + k + 256, 0, 0);   // global_prefetch_b8, ~1KB ahead
#pragma unroll
    for (int j = 0; j < 4; ++j) {
      const int base = k + 4 * j;
      v2f a  = *(const v2f*)(rowp + base);      // 8B-aligned b64 load
      v2f bb = *(const v2f*)(ep + base);        // embedding chunk (L0-resident)
      acc = __builtin_amdgcn_wmma_f32_16x16x4_f32(
          /*neg_a=*/false, a, /*neg_b=*/false, bb,
          /*c_mod=*/(short)0, acc, /*reuse_a=*/false, /*reuse_b=*/false);
    }
  }

  if (mrow == 0) {                        // lanes 0 (M=0..7) and 16 (M=8..15)
#pragma unroll
    for (int j = 0; j < 8; ++j) wdot[r0 + half * 8 + j] = acc[j];
  }
}

// ---------------------------------------------------------------------------
// Kernel 2: per batch, exact 409th-largest of key = gumbel_noise - w via
// 32-bit MSB radix-select on order-preserving uint keys in LDS.
// mask[b][s] = (key >= kth_largest) ? 0 : 1   (dropped positions -> 0)
// ---------------------------------------------------------------------------
__global__ __launch_bounds__(256) void select_mask_kernel(
    const float* __restrict__ wdot, const float* __restrict__ gumbel_u,
    float* __restrict__ maskbuf) {
  __shared__ unsigned int ukey[SEQ_N];
  __shared__ unsigned int cnt;
  const int b = blockIdx.x;
  const int tid = threadIdx.x;

  for (int i = tid; i < SEQ_N; i += 256) {
    const float wv = wdot[(size_t)i * BS_N + b];        // w[b][s=i]
    const float u  = gumbel_u[b * SEQ_N + i];
    const float g  = -logf(-logf(u + 1e-20f) + 1e-20f); // gumbel noise
    const float key = g - wv;  // order-equivalent to log_softmax(mx-w)+gumbel
    unsigned int uk = __float_as_uint(key);
    uk = (uk & 0x80000000u) ? ~uk : (uk | 0x80000000u); // monotone float->uint
    ukey[i] = uk;
  }
  __syncthreads();

  unsigned int prefix = 0;
  int kk = NDROP;
  for (int bit = 31; bit >= 0; --bit) {
    const unsigned int tb = 1u << bit;
    const unsigned int hm = ~((tb << 1) - 1u);  // bits above `bit` (0 at bit 31)
    if (tid == 0) cnt = 0;
    __syncthreads();
    unsigned int local = 0;
    for (int i = tid; i < SEQ_N; i += 256) {
      const unsigned int uk = ukey[i];
      if (((uk & hm) == (prefix & hm)) && (uk & tb)) ++local;
    }
    atomicAdd(&cnt, local);
    __syncthreads();
    const unsigned int c = cnt;
    if ((int)c >= kk) prefix |= tb; else kk -= (int)c;
    __syncthreads();
  }
  // prefix == exact value of the NDROP-th largest key
  for (int i = tid; i < SEQ_N; i += 256)
    maskbuf[b * SEQ_N + i] = (ukey[i] >= prefix) ? 0.0f : 1.0f;
}

// ---------------------------------------------------------------------------
// Kernel 3: out = LayerNorm(x + h*mask) * gamma + beta, one block per row.
// 256 threads x float4 (b128 loads); wave32 shfl_xor + LDS reduction.
// h is L2-resident from kernel 1 (64MB << 192MB L2), so HBM traffic is
// x + out only on this pass.
// ---------------------------------------------------------------------------
__global__ __launch_bounds__(256) void fused_residual_ln(
    const float4* __restrict__ x, const float4* __restrict__ h,
    const float4* __restrict__ gamma, const float4* __restrict__ beta,
    const float* __restrict__ maskbuf, float4* __restrict__ out) {
  const int row = blockIdx.x;          // row = s*BS + b
  const int b = row & (BS_N - 1);
  const int s = row >> 2;
  const float m = maskbuf[b * SEQ_N + s];
  const int tid = threadIdx.x;
  const size_t base = (size_t)row * (SIZE_D / 4);

  const float4 xv = x[base + tid];
  const float4 hv = h[base + tid];
  float4 y;
  y.x = xv.x + hv.x * m;
  y.y = xv.y + hv.y * m;
  y.z = xv.z + hv.z * m;
  y.w = xv.w + hv.w * m;

  float sum = y.x + y.y + y.z + y.w;
  float sq  = y.x * y.x + y.y * y.y + y.z * y.z + y.w * y.w;
#pragma unroll
  for (int off = 16; off > 0; off >>= 1) {   // wave32 butterfly reduce
    sum += __shfl_xor(sum, off, 32);
    sq  += __shfl_xor(sq,  off, 32);
  }
  __shared__ float s_sum[8], s_sq[8];
  const int wv = tid >> 5;
  if ((tid & 31) == 0) { s_sum[wv] = sum; s_sq[wv] = sq; }
  __syncthreads();
  float ts = 0.f, tq = 0.f;
#pragma unroll
  for (int i = 0; i < 8; ++i) { ts += s_sum[i]; tq += s_sq[i]; }

  const float mu   = ts * (1.0f / SIZE_D);
  const float var  = tq * (1.0f / SIZE_D) - mu * mu;
  const float rinv = rsqrtf(var + 1e-12f);

  const float4 g  = gamma[tid];
  const float4 bt = beta[tid];
  float4 o;
  o.x = (y.x - mu) * rinv * g.x + bt.x;
  o.y = (y.y - mu) * rinv * g.y + bt.y;
  o.z = (y.z - mu) * rinv * g.z + bt.z;
  o.w = (y.w - mu) * rinv * g.w + bt.w;
  out[base + tid] = o;
}

extern "C" void kernel_launch(void* const* d_in, const int* in_sizes, int n_in,
                              void* d_out, int out_size, void* d_ws, size_t ws_size,
                              hipStream_t stream) {
  const float* x        = (const float*)d_in[0];  // (SEQ, BS, SIZE)
  const float* h        = (const float*)d_in[1];  // (SEQ, BS, SIZE)
  const float* e        = (const float*)d_in[2];  // (SIZE,)
  const float* gamma    = (const float*)d_in[3];  // (SIZE,)
  const float* beta     = (const float*)d_in[4];  // (SIZE,)
  const float* gumbel_u = (const float*)d_in[5];  // (BS, SEQ)
  // d_in[6] = lengths: unused (also unused by the reference computation)

  float* wdot    = (float*)d_ws;       // ROWS floats (64 KB)
  float* maskbuf = wdot + ROWS;        // BS*SEQ floats (64 KB)

  task_attn_gemv_wmma<<<ROWS / (16 * 8), 256, 0, stream>>>(h, e, wdot);
  select_mask_kernel<<<BS_N, 256, 0, stream>>>(wdot, gumbel_u, maskbuf);
  fused_residual_ln<<<ROWS, 256, 0, stream>>>(
      (const float4*)x, (const float4*)h, (const float4*)gamma,
      (const float4*)beta, maskbuf, (float4*)d_out);
}